// LinearBasis_85220741087482
// MI455X (gfx1250) — compile-verified
//
#include <hip/hip_runtime.h>
#include <hip/hip_bf16.h>

typedef float v2f __attribute__((ext_vector_type(2)));
typedef float v8f __attribute__((ext_vector_type(8)));

#define LAT 256          // lattice side
#define NT  512          // threads per block (16 waves, wave32)
#define RPI 8            // rows streamed per iteration
#define RING 32          // ring buffer rows (power of 2, >= 3*RPI+1 for 1-barrier safety)

__global__ __launch_bounds__(NT)
void ising_obs_kernel(const float* __restrict__ x, float* __restrict__ out) {
    __shared__ float ring[RING * LAT];     // 32 KB row ring; reused as reduction scratch
    const int b   = blockIdx.x;
    const int tid = threadIdx.x;
    const int c4  = (tid & 63) << 2;       // base column of this thread's 4 elements
    const int rr  = tid >> 6;              // row offset within the 8-row slab
    const float* __restrict__ img = x + (size_t)b * (LAT * LAT);

    // Prologue: stage periodic row (L-1), the "up" neighbor of row 0, into slot 31.
    if (tid < 64) {
        const float4 v = *(const float4*)(img + (LAT - 1) * LAT + tid * 4);
        *(float4*)(&ring[(RING - 1) * LAT + tid * 4]) = v;
    }

    float a0 = 0.f, a1 = 0.f, a2 = 0.f;

    for (int i = 0; i < LAT; i += RPI) {
        const int r = i + rr;
        // coalesced slab load: flat offset i*256 + tid*4 == r*LAT + c4
        const float4 v = *(const float4*)(img + (size_t)i * LAT + tid * 4);
        *(float4*)(&ring[(r & (RING - 1)) * LAT + c4]) = v;
        if (i + RPI < LAT)
            __builtin_prefetch(img + (size_t)(i + RPI) * LAT + tid * 4, 0, 1);
        __syncthreads();

        const int sc   = r & (RING - 1);
        const int su   = (r - 1) & (RING - 1);
        const int lcol = (c4 + LAT - 1) & (LAT - 1);
        const float4 up = *(const float4*)(&ring[su * LAT + c4]);
        const float  l  = ring[sc * LAT + lcol];   // left neighbor of element 0
        const float  ul = ring[su * LAT + lcol];   // up-left neighbor of element 0

        // nearest: cur*up + cur*left
        a0 += v.x * up.x + v.y * up.y + v.z * up.z + v.w * up.w;
        a0 += v.x * l    + v.y * v.x  + v.z * v.y  + v.w * v.z;
        // next-nearest: cur*upleft + up*left
        a1 += v.x * ul   + v.y * up.x + v.z * up.y + v.w * up.z;
        a1 += up.x * l   + up.y * v.x + up.z * v.y + up.w * v.z;
        // plaquette: cur*up*left*upleft
        a2 += (v.x * up.x) * (l    * ul);
        a2 += (v.y * up.y) * (v.x * up.x);
        a2 += (v.z * up.z) * (v.y * up.y);
        a2 += (v.w * up.w) * (v.z * up.z);
    }

    __syncthreads();                       // ring reads done; reuse LDS as scratch
    ring[tid]          = a0;
    ring[NT + tid]     = a1;
    ring[2 * NT + tid] = a2;
    __syncthreads();

    // Wave 0 reduces 512 partials per quantity with chained f32 WMMAs.
    // A = ones(16x4), B = 64 partials/chunk. Summing ALL of D counts every B
    // element exactly 16 times regardless of operand lane layout -> *1/16.
    if (tid < 32) {
        float res[3];
        const v2f aones = {1.0f, 1.0f};
        #pragma unroll
        for (int q = 0; q < 3; ++q) {
            const float* sq = &ring[q * NT];
            v8f acc = {0.f, 0.f, 0.f, 0.f, 0.f, 0.f, 0.f, 0.f};
            #pragma unroll
            for (int chunk = 0; chunk < NT / 64; ++chunk) {
                v2f bv;
                bv.x = sq[chunk * 64 + 2 * tid];
                bv.y = sq[chunk * 64 + 2 * tid + 1];
                acc = __builtin_amdgcn_wmma_f32_16x16x4_f32(
                          false, aones, false, bv, (short)0, acc, false, false);
            }
            float s = acc[0] + acc[1] + acc[2] + acc[3]
                    + acc[4] + acc[5] + acc[6] + acc[7];
            #pragma unroll
            for (int off = 16; off > 0; off >>= 1)
                s += __shfl_xor(s, off, 32);
            res[q] = s * 0.0625f;
        }
        if (tid == 0) {
            out[b * 3 + 0] = res[0];
            out[b * 3 + 1] = res[1];
            out[b * 3 + 2] = res[2];
        }
    }
}

extern "C" void kernel_launch(void* const* d_in, const int* in_sizes, int n_in,
                              void* d_out, int out_size, void* d_ws, size_t ws_size,
                              hipStream_t stream) {
    const float* x = (const float*)d_in[0];
    float* out = (float*)d_out;
    const int B = in_sizes[0] / (LAT * LAT);   // 1024
    ising_obs_kernel<<<B, NT, 0, stream>>>(x, out);
}